// SDALayer_17875653886323
// MI455X (gfx1250) — compile-verified
//
#include <hip/hip_runtime.h>
#include <math.h>

typedef __attribute__((ext_vector_type(2))) float v2f;
typedef __attribute__((ext_vector_type(8))) float v8f;

#define B_  4
#define H_  12
#define S_  2048
#define D_  64
#define QT  16     // query rows per workgroup
#define NW  4      // waves per workgroup (wave32)
#define QPAD (D_ + 4)   // LDS row stride for Q tile (conflict-free, 16B aligned)
#define PPAD 18         // LDS row stride for P tile (conflict-free, 8B aligned)

__global__ __launch_bounds__(NW * 32) void sdpa_fwd_f32_wmma(
    const float* __restrict__ qg,
    const float* __restrict__ kg,
    const float* __restrict__ vg,
    const unsigned char* __restrict__ maskg,   // [B,1,1,S] bool
    float* __restrict__ out_vals,              // [B,H,S,D]
    float* __restrict__ out_logits)            // [B,H,S,S]
{
    __shared__ float lds_q[QT][QPAD];
    __shared__ float lds_p[NW][QT][PPAD];
    __shared__ float lds_o[NW][QT][D_];
    __shared__ float lds_m[NW][QT];
    __shared__ float lds_l[NW][QT];

    const int tid  = threadIdx.x;
    const int wave = tid >> 5;
    const int lane = tid & 31;
    const int half = lane >> 4;      // which 16-lane group
    const int l16  = lane & 15;

    const int blocks_per_bh = S_ / QT;                 // 128
    const int bh = blockIdx.x / blocks_per_bh;         // 0..B*H-1
    const int qb = blockIdx.x % blocks_per_bh;         // query tile index
    const int b  = bh / H_;

    const float* qbase = qg + ((size_t)bh * S_ + (size_t)qb * QT) * D_;
    const float* kbase = kg + (size_t)bh * S_ * D_;
    const float* vbase = vg + (size_t)bh * S_ * D_;
    const unsigned char* mbase = maskg + (size_t)b * S_;
    float* logit_base = out_logits + ((size_t)bh * S_ + (size_t)qb * QT) * S_;
    float* val_base   = out_vals   + ((size_t)bh * S_ + (size_t)qb * QT) * D_;

    // ---- stage Q tile (16x64 f32) into LDS, b128 loads, coalesced ----
    {
        const float4* src = (const float4*)qbase;
        for (int i = tid; i < QT * D_ / 4; i += blockDim.x) {
            float4 val = src[i];
            int row = (i * 4) / D_;
            int col = (i * 4) % D_;
            *(float4*)&lds_q[row][col] = val;
        }
    }
    __syncthreads();

    const float scale = 0.125f;    // 1/sqrt(64)
    const float NEG_INF = -__builtin_inff();

    // per-lane softmax state: 8 rows (r + 8*half) of the 16-row query tile
    float m_r[8], l_r[8];
    v8f o_acc[4];
    #pragma unroll
    for (int r = 0; r < 8; ++r) { m_r[r] = NEG_INF; l_r[r] = 0.0f; }
    #pragma unroll
    for (int db = 0; db < 4; ++db) o_acc[db] = (v8f){0,0,0,0,0,0,0,0};

    // each wave walks a disjoint set of 16-wide key blocks
    for (int n0 = wave * 16; n0 < S_; n0 += NW * 16) {
        // ---- S tile = Q (16xD) * K^T (Dx16) via 16 fp32 WMMAs ----
        v8f c = (v8f){0,0,0,0,0,0,0,0};
        const float* krow = kbase + (size_t)(n0 + l16) * D_;  // lane's key row
        #pragma unroll
        for (int kk = 0; kk < D_; kk += 4) {
            // A (16x4): lane = row m=l16; VGPR pair = K = kk+2*half, +1
            v2f a = *(const v2f*)&lds_q[l16][kk + 2 * half];
            // B (4x16): lane = col n=l16; VGPR pair = K = kk+2*half, +1
            v2f bv = *(const v2f*)&krow[kk + 2 * half];
            c = __builtin_amdgcn_wmma_f32_16x16x4_f32(
                    false, a, false, bv, (short)0, c, false, false);
        }

        // scale + key-padding mask (column n0+l16 masked for all rows)
        const bool masked = mbase[n0 + l16] != 0;
        #pragma unroll
        for (int r = 0; r < 8; ++r) {
            float x = c[r] * scale;
            c[r] = masked ? NEG_INF : x;
        }

        // ---- write logits tile (output #2), coalesced per 16-lane group ----
        {
            float* lrow = logit_base + n0 + l16;
            #pragma unroll
            for (int r = 0; r < 8; ++r)
                lrow[(size_t)(r + 8 * half) * S_] = c[r];
        }

        // ---- online softmax update (row reductions across 16-lane groups) --
        float alpha[8];
        #pragma unroll
        for (int r = 0; r < 8; ++r) {
            float x = c[r];
            x = fmaxf(x, __shfl_xor(x, 1));
            x = fmaxf(x, __shfl_xor(x, 2));
            x = fmaxf(x, __shfl_xor(x, 4));
            x = fmaxf(x, __shfl_xor(x, 8));
            float mnew = fmaxf(m_r[r], x);
            alpha[r] = __expf(m_r[r] - mnew);
            float p = __expf(c[r] - mnew);
            c[r] = p;
            float s = p;
            s += __shfl_xor(s, 1);
            s += __shfl_xor(s, 2);
            s += __shfl_xor(s, 4);
            s += __shfl_xor(s, 8);
            l_r[r] = l_r[r] * alpha[r] + s;
            m_r[r] = mnew;
        }
        #pragma unroll
        for (int db = 0; db < 4; ++db)
            #pragma unroll
            for (int r = 0; r < 8; ++r)
                o_acc[db][r] *= alpha[r];

        // ---- stage P (C layout -> A layout) through per-wave LDS tile ----
        #pragma unroll
        for (int r = 0; r < 8; ++r)
            lds_p[wave][r + 8 * half][l16] = c[r];
        // same-wave DS ordering; compiler inserts s_wait_dscnt for the RAW

        // ---- O += P (16x16) * V (16x64): 4 K-steps x 4 D-blocks of WMMA ----
        #pragma unroll
        for (int kk2 = 0; kk2 < 16; kk2 += 4) {
            v2f pa = *(const v2f*)&lds_p[wave][l16][kk2 + 2 * half];
            const float* vrow = vbase + (size_t)(n0 + kk2 + 2 * half) * D_;
            #pragma unroll
            for (int db = 0; db < 4; ++db) {
                v2f bv;
                bv[0] = vrow[db * 16 + l16];
                bv[1] = vrow[D_ + db * 16 + l16];
                o_acc[db] = __builtin_amdgcn_wmma_f32_16x16x4_f32(
                                false, pa, false, bv, (short)0, o_acc[db],
                                false, false);
            }
        }
    }

    // ---- publish per-wave partials (m, l, O) to LDS ----
    #pragma unroll
    for (int r = 0; r < 8; ++r) {
        const int row = r + 8 * half;
        #pragma unroll
        for (int db = 0; db < 4; ++db)
            lds_o[wave][row][db * 16 + l16] = o_acc[db][r];
    }
    if (l16 == 0) {
        #pragma unroll
        for (int r = 0; r < 8; ++r) {
            lds_m[wave][r + 8 * half] = m_r[r];
            lds_l[wave][r + 8 * half] = l_r[r];
        }
    }
    __syncthreads();

    // ---- cross-wave flash combine + normalize + write values (output #1) --
    for (int i = tid; i < QT * D_; i += blockDim.x) {
        const int row = i / D_;
        const int d   = i % D_;
        float M = lds_m[0][row];
        #pragma unroll
        for (int w = 1; w < NW; ++w) M = fmaxf(M, lds_m[w][row]);
        float L = 0.0f, O = 0.0f;
        #pragma unroll
        for (int w = 0; w < NW; ++w) {
            float sc = __expf(lds_m[w][row] - M);
            L += lds_l[w][row] * sc;
            O += lds_o[w][row][d] * sc;
        }
        val_base[(size_t)row * D_ + d] = O / L;
    }
}

extern "C" void kernel_launch(void* const* d_in, const int* in_sizes, int n_in,
                              void* d_out, int out_size, void* d_ws, size_t ws_size,
                              hipStream_t stream) {
    const float* q = (const float*)d_in[0];
    const float* k = (const float*)d_in[1];
    const float* v = (const float*)d_in[2];
    const unsigned char* mask = (const unsigned char*)d_in[3];  // bool [B,1,1,S]

    float* out_vals   = (float*)d_out;                             // [B,H,S,D]
    float* out_logits = (float*)d_out + (size_t)B_ * H_ * S_ * D_; // [B,H,S,S]

    const int grid = B_ * H_ * (S_ / QT);   // 6144 workgroups
    sdpa_fwd_f32_wmma<<<grid, NW * 32, 0, stream>>>(q, k, v, mask,
                                                    out_vals, out_logits);
}